// MaskDecoder_67869073211799
// MI455X (gfx1250) — compile-verified
//
#include <hip/hip_runtime.h>
#include <hip/hip_bf16.h>
#include <math.h>

// ---------------- problem constants ----------------
#define BN   32
#define CN   256
#define HWN  4096      // 64*64
#define TN   13        // 1 iou + 4 mask + 8 sparse tokens
#define MN   4
#define EN   4
#define SEN  2
#define C4N  64
#define C8N  32
#define HW2N 16384     // 128*128
#define HW4N 65536     // 256*256
#define SCALE 0.0625f  // 1/sqrt(256)

// ---------------- workspace layout (floats) ----------------
#define WS_TOK   0u                         // B*13*256
#define WS_QH    106496u                    // B*13*256 (pre-scaled)
#define WS_U     212992u                    // B*13*256
#define WS_HS    319488u                    // B*13*256
#define WS_V2    425984u                    // B*13*256
#define WS_W2T   532480u                    // B*256*13 (pre-scaled)
#define WS_HYP   638976u                    // B*4*32
#define WS_ATT   643072u                    // B*13*4096
#define WS_BIGA  2359296u                   // src2 (NCHW)  -> later X2 [(b*16384+p2)*64+o]
#define WS_BIGB  (WS_BIGA + 33554432u)      // y  [row*256+n]

typedef float v2f __attribute__((ext_vector_type(2)));
typedef float v8f __attribute__((ext_vector_type(8)));

#if defined(__has_builtin)
#if __has_builtin(__builtin_amdgcn_wmma_f32_16x16x4_f32)
#define HAVE_WMMA_F32X4 1
#endif
#endif

__device__ __forceinline__ float gelu_exact(float x) {
  return 0.5f * x * (1.0f + erff(x * 0.70710678118654752f));
}

// ---------------- K0: tokens + Qh = (tok@wq)@wk^T * scale ----------------
__global__ __launch_bounds__(256) void k0_tokens_qh(
    const float* __restrict__ iou_token, const float* __restrict__ mask_tokens,
    const float* __restrict__ sparse, const float* __restrict__ wq,
    const float* __restrict__ wk, float* __restrict__ tokens_g,
    float* __restrict__ qh_g) {
  int t = blockIdx.x, b = blockIdx.y, c = threadIdx.x;
  __shared__ float tok[CN];
  __shared__ float q[CN];
  float tv;
  if (t == 0)      tv = iou_token[c];
  else if (t < 5)  tv = mask_tokens[(t - 1) * CN + c];
  else             tv = sparse[((b * 8) + (t - 5)) * CN + c];
  tok[c] = tv;
  tokens_g[(b * TN + t) * CN + c] = tv;
  __syncthreads();
  float qa = 0.f;
  for (int cc = 0; cc < CN; ++cc) qa += tok[cc] * wq[cc * CN + c];
  q[c] = qa;
  __syncthreads();
  float qk = 0.f;
  for (int j = 0; j < CN; ++j) qk += q[j] * wk[c * CN + j];
  qh_g[(b * TN + t) * CN + c] = qk * SCALE;
}

// ---------------- K1: S1[b,t,n] = Qh[b,t,:] . kmem[b,n,:] ----------------
__global__ __launch_bounds__(256) void k1_scores1(
    const float* __restrict__ img, const float* __restrict__ dense,
    const float* __restrict__ pe, const float* __restrict__ qh_g,
    float* __restrict__ attn) {
  int b = blockIdx.y, tid = threadIdx.x;
  int n = blockIdx.x * 256 + tid;
  __shared__ float qh[CN * 16];  // [c][t], pad 16
  for (int t = 0; t < TN; ++t) qh[tid * 16 + t] = qh_g[(b * TN + t) * CN + tid];
  __syncthreads();
  float s[TN];
  for (int t = 0; t < TN; ++t) s[t] = 0.f;
  const float* ib = img + (size_t)b * CN * HWN + n;
  const float* db = dense + (size_t)b * CN * HWN + n;
  const float* pb = pe + n;
  for (int c = 0; c < CN; ++c) {
    float kv = ib[c * HWN] + db[c * HWN] + pb[c * HWN];
    const float4 q0 = *(const float4*)&qh[c * 16 + 0];
    const float4 q1 = *(const float4*)&qh[c * 16 + 4];
    const float4 q2 = *(const float4*)&qh[c * 16 + 8];
    const float q3 = qh[c * 16 + 12];
    s[0] += q0.x * kv; s[1] += q0.y * kv; s[2] += q0.z * kv; s[3] += q0.w * kv;
    s[4] += q1.x * kv; s[5] += q1.y * kv; s[6] += q1.z * kv; s[7] += q1.w * kv;
    s[8] += q2.x * kv; s[9] += q2.y * kv; s[10] += q2.z * kv; s[11] += q2.w * kv;
    s[12] += q3 * kv;
  }
  for (int t = 0; t < TN; ++t)
    attn[((size_t)(b * TN + t)) * HWN + n] = s[t];
}

// ---------------- K2: softmax over n (4096) per (b,t) row ----------------
__global__ __launch_bounds__(256) void k2_softmax_rows(float* __restrict__ attn) {
  int row = blockIdx.x, tid = threadIdx.x;
  size_t base = (size_t)row * HWN;
  __shared__ float red[256];
  float r[16];
  float mx = -3.402823466e38f;
  for (int k = 0; k < 16; ++k) {
    r[k] = attn[base + tid + k * 256];
    mx = fmaxf(mx, r[k]);
  }
  red[tid] = mx; __syncthreads();
  for (int off = 128; off >= 1; off >>= 1) {
    if (tid < off) red[tid] = fmaxf(red[tid], red[tid + off]);
    __syncthreads();
  }
  mx = red[0]; __syncthreads();
  float sum = 0.f;
  for (int k = 0; k < 16; ++k) { r[k] = expf(r[k] - mx); sum += r[k]; }
  red[tid] = sum; __syncthreads();
  for (int off = 128; off >= 1; off >>= 1) {
    if (tid < off) red[tid] += red[tid + off];
    __syncthreads();
  }
  float inv = 1.0f / red[0];
  for (int k = 0; k < 16; ++k) attn[base + tid + k * 256] = r[k] * inv;
}

// ---------------- K3: U[b,t,c] = sum_n attn[b,t,n]*kmem[b,n,c] ----------------
__global__ __launch_bounds__(256) void k3_attn_kmem(
    const float* __restrict__ img, const float* __restrict__ dense,
    const float* __restrict__ pe, const float* __restrict__ attn,
    float* __restrict__ u_g) {
  int b = blockIdx.x >> 8, c = blockIdx.x & 255, tid = threadIdx.x;
  float part[TN];
  for (int t = 0; t < TN; ++t) part[t] = 0.f;
  size_t obase = ((size_t)b * CN + c) * HWN;
  for (int k = 0; k < 16; ++k) {
    int n = k * 256 + tid;
    float kv = img[obase + n] + dense[obase + n] + pe[(size_t)c * HWN + n];
    for (int t = 0; t < TN; ++t)
      part[t] += attn[((size_t)(b * TN + t)) * HWN + n] * kv;
  }
  __shared__ float red[TN * 256];
  for (int t = 0; t < TN; ++t) red[t * 256 + tid] = part[t];
  __syncthreads();
  for (int off = 128; off >= 1; off >>= 1) {
    if (tid < off)
      for (int t = 0; t < TN; ++t) red[t * 256 + tid] += red[t * 256 + tid + off];
    __syncthreads();
  }
  if (tid < TN) u_g[(b * TN + tid) * CN + c] = red[tid * 256];
}

// ---------------- K4: hs = tokens + U @ wv ----------------
__global__ __launch_bounds__(256) void k4_hs(
    const float* __restrict__ u_g, const float* __restrict__ wv,
    const float* __restrict__ tokens_g, float* __restrict__ hs_g) {
  int t = blockIdx.x, b = blockIdx.y, c = threadIdx.x;
  __shared__ float ul[CN];
  ul[c] = u_g[(b * TN + t) * CN + c];
  __syncthreads();
  float acc = tokens_g[(b * TN + t) * CN + c];
  for (int j = 0; j < CN; ++j) acc += ul[j] * wv[j * CN + c];
  hs_g[(b * TN + t) * CN + c] = acc;
}

// ---------------- K5: k2=hs@wk2, v2=hs@wv2, W2t[c][t]=wq2[c,:].k2[t]*scale ----------------
__global__ __launch_bounds__(256) void k5_proj2(
    const float* __restrict__ hs_g, const float* __restrict__ wk2,
    const float* __restrict__ wv2, const float* __restrict__ wq2,
    float* __restrict__ v2_g, float* __restrict__ w2t_g) {
  int b = blockIdx.x, tid = threadIdx.x;
  __shared__ float hsl[TN * CN];
  __shared__ float k2l[TN * CN];
  for (int t = 0; t < TN; ++t) hsl[t * CN + tid] = hs_g[(b * TN + t) * CN + tid];
  __syncthreads();
  for (int t = 0; t < TN; ++t) {
    float a1 = 0.f, a2 = 0.f;
    for (int j = 0; j < CN; ++j) {
      float h = hsl[t * CN + j];
      a1 += h * wk2[j * CN + tid];
      a2 += h * wv2[j * CN + tid];
    }
    k2l[t * CN + tid] = a1;
    v2_g[(b * TN + t) * CN + tid] = a2;
  }
  __syncthreads();
  for (int t = 0; t < TN; ++t) {
    float a = 0.f;
    for (int j = 0; j < CN; ++j) a += wq2[tid * CN + j] * k2l[t * CN + j];
    w2t_g[((b << 8) + tid) * TN + t] = a * SCALE;
  }
}

// ---------------- K6: S2 + softmax(13) + src2(NCHW) = src + attn2@v2 ----------------
__global__ __launch_bounds__(256) void k6_attn2_src2(
    const float* __restrict__ img, const float* __restrict__ dense,
    const float* __restrict__ pe, const float* __restrict__ w2t_g,
    const float* __restrict__ v2_g, float* __restrict__ src2) {
  int b = blockIdx.y, tid = threadIdx.x;
  int n = blockIdx.x * 256 + tid;
  __shared__ float w2l[CN * 16];  // [c][t]
  __shared__ float v2l[CN * 16];  // [c][t]
  for (int t = 0; t < TN; ++t) {
    w2l[tid * 16 + t] = w2t_g[((b << 8) + tid) * TN + t];
    v2l[tid * 16 + t] = v2_g[(b * TN + t) * CN + tid];
  }
  __syncthreads();
  float s[TN];
  for (int t = 0; t < TN; ++t) s[t] = 0.f;
  const float* ib = img + (size_t)b * CN * HWN + n;
  const float* db = dense + (size_t)b * CN * HWN + n;
  const float* pb = pe + n;
  for (int c = 0; c < CN; ++c) {
    float kv = ib[c * HWN] + db[c * HWN] + pb[c * HWN];
    const float4 q0 = *(const float4*)&w2l[c * 16 + 0];
    const float4 q1 = *(const float4*)&w2l[c * 16 + 4];
    const float4 q2 = *(const float4*)&w2l[c * 16 + 8];
    const float q3 = w2l[c * 16 + 12];
    s[0] += q0.x * kv; s[1] += q0.y * kv; s[2] += q0.z * kv; s[3] += q0.w * kv;
    s[4] += q1.x * kv; s[5] += q1.y * kv; s[6] += q1.z * kv; s[7] += q1.w * kv;
    s[8] += q2.x * kv; s[9] += q2.y * kv; s[10] += q2.z * kv; s[11] += q2.w * kv;
    s[12] += q3 * kv;
  }
  float mx = s[0];
  for (int t = 1; t < TN; ++t) mx = fmaxf(mx, s[t]);
  float sum = 0.f;
  for (int t = 0; t < TN; ++t) { s[t] = expf(s[t] - mx); sum += s[t]; }
  float inv = 1.0f / sum;
  for (int t = 0; t < TN; ++t) s[t] *= inv;
  float* ob = src2 + (size_t)b * CN * HWN + n;
  for (int c = 0; c < CN; ++c) {
    float acc = ib[c * HWN] + db[c * HWN];  // residual: src without pos
    const float4 v0 = *(const float4*)&v2l[c * 16 + 0];
    const float4 v1 = *(const float4*)&v2l[c * 16 + 4];
    const float4 v2 = *(const float4*)&v2l[c * 16 + 8];
    const float v3 = v2l[c * 16 + 12];
    acc += s[0] * v0.x + s[1] * v0.y + s[2] * v0.z + s[3] * v0.w;
    acc += s[4] * v1.x + s[5] * v1.y + s[6] * v1.z + s[7] * v1.w;
    acc += s[8] * v2.x + s[9] * v2.y + s[10] * v2.z + s[11] * v2.w;
    acc += s[12] * v3;
    ob[c * HWN] = acc;
  }
}

// ---------------- K7: up1 GEMM via FP32 WMMA: y[row][n] = sum_c X[row][c]*w1[c][n] ----------------
// X[row=(b*4096+p)][c] = src2[b][c][p]; w1 is up1_w viewed as 256x256 (n = o*4+dh*2+dw)
__global__ __launch_bounds__(256) void k7_up1_gemm(
    const float* __restrict__ src2, const float* __restrict__ w1,
    float* __restrict__ y) {
  __shared__ float At[16 * 260];
  int blk = blockIdx.x;           // 8192 row-tiles
  int b = blk >> 8;
  int p0 = (blk & 255) << 4;
  int tid = threadIdx.x;
  {
    const float* sp = src2 + ((size_t)(b * CN + tid)) * HWN + p0;
    float4 a0 = *(const float4*)(sp + 0);
    float4 a1 = *(const float4*)(sp + 4);
    float4 a2 = *(const float4*)(sp + 8);
    float4 a3 = *(const float4*)(sp + 12);
    float v[16] = {a0.x, a0.y, a0.z, a0.w, a1.x, a1.y, a1.z, a1.w,
                   a2.x, a2.y, a2.z, a2.w, a3.x, a3.y, a3.z, a3.w};
    for (int p = 0; p < 16; ++p) At[p * 260 + tid] = v[p];
  }
  __syncthreads();
  int rowbase = blk * 16;
#ifdef HAVE_WMMA_F32X4
  int wave = tid >> 5, lane = tid & 31;
  int half = lane >> 4, lrow = lane & 15;
  v8f acc0 = {0.f, 0.f, 0.f, 0.f, 0.f, 0.f, 0.f, 0.f};
  v8f acc1 = acc0;
  int n0 = (wave * 2) * 16 + lrow, n1 = n0 + 16;
  for (int kk = 0; kk < CN; kk += 4) {
    int ki = kk + 2 * half;
    v2f a = *(const v2f*)&At[lrow * 260 + ki];
    v2f b0, b1;
    b0.x = w1[ki * CN + n0];       b0.y = w1[(ki + 1) * CN + n0];
    b1.x = w1[ki * CN + n1];       b1.y = w1[(ki + 1) * CN + n1];
    acc0 = __builtin_amdgcn_wmma_f32_16x16x4_f32(false, a, false, b0, (short)0,
                                                 acc0, false, false);
    acc1 = __builtin_amdgcn_wmma_f32_16x16x4_f32(false, a, false, b1, (short)0,
                                                 acc1, false, false);
  }
  for (int r = 0; r < 8; ++r) {
    int mrow = r + 8 * half;
    y[(size_t)(rowbase + mrow) * CN + n0] = acc0[r];
    y[(size_t)(rowbase + mrow) * CN + n1] = acc1[r];
  }
#else
  int n = tid;
  for (int r = 0; r < 16; ++r) {
    float acc = 0.f;
    for (int c = 0; c < CN; ++c) acc += At[r * 260 + c] * w1[c * CN + n];
    y[(size_t)(rowbase + r) * CN + n] = acc;
  }
#endif
}

// ---------------- K8: bias + LayerNorm(64ch per tap) + GELU -> X2[(b*16384+p2)*64+o] ----------------
__global__ __launch_bounds__(256) void k8_ln_gelu(
    const float* __restrict__ y, const float* __restrict__ b1,
    const float* __restrict__ lnw, const float* __restrict__ lnb,
    float* __restrict__ X2) {
  int row = blockIdx.x;               // b*4096 + p
  int b = row >> 12, p = row & 4095;
  int h = p >> 6, w = p & 63;
  int t = threadIdx.x;
  int g = t & 3, o = t >> 2;          // tap, channel
  float val = y[(size_t)row * CN + t] + b1[o];
  __shared__ float s1[256];
  __shared__ float s2[256];
  s1[g * 64 + o] = val;
  s2[g * 64 + o] = val * val;
  __syncthreads();
  for (int off = 32; off >= 1; off >>= 1) {
    if (o < off) {
      s1[g * 64 + o] += s1[g * 64 + o + off];
      s2[g * 64 + o] += s2[g * 64 + o + off];
    }
    __syncthreads();
  }
  float mu = s1[g * 64] * (1.0f / 64.0f);
  float var = s2[g * 64] * (1.0f / 64.0f) - mu * mu;
  var = fmaxf(var, 0.f) + 1e-6f;
  float norm = (val - mu) / sqrtf(var);
  float res = lnw[o] * norm + lnb[o];
  float gel = gelu_exact(res);
  int dh = g >> 1, dw = g & 1;
  int p2 = (2 * h + dh) * 128 + (2 * w + dw);
  X2[((size_t)(b * HW2N + p2)) * C4N + o] = gel;
}

// ---------------- K9: up2 conv-T GEMM via FP32 WMMA + GELU + masks ----------------
// A = X2 rows (16 pixels x 64ch), B = up2_w as 64x128 (n = oc*4 + dh*2 + dw).
// D fragments get bias+GELU in-register, staged to LDS, then contracted with
// hyper_in (32-deep) and scattered into the 256x256 mask planes.
__global__ __launch_bounds__(256) void k9_up2_masks(
    const float* __restrict__ X2, const float* __restrict__ w2g,
    const float* __restrict__ b2g, const float* __restrict__ hyperin,
    const int* __restrict__ mmflag, float* __restrict__ out) {
  __shared__ float At[16 * 68];        // 16 pixel-rows x 64 ch (padded)
  __shared__ float G[16 * 132];        // gelu'd up2 tile: 16 rows x 128 (padded)
  __shared__ float hyp_l[MN * C8N];
  int blk = blockIdx.x;                // 32768 = B * 1024
  int b = blk >> 10;
  int p2base = (blk & 1023) << 4;
  int tid = threadIdx.x;
  {
    int r = tid >> 4, c4 = (tid & 15) * 4;
    const float4 t4 =
        *(const float4*)(X2 + ((size_t)(b * HW2N + p2base + r)) * C4N + c4);
    At[r * 68 + c4 + 0] = t4.x; At[r * 68 + c4 + 1] = t4.y;
    At[r * 68 + c4 + 2] = t4.z; At[r * 68 + c4 + 3] = t4.w;
  }
  if (tid < MN * C8N) hyp_l[tid] = hyperin[b * MN * C8N + tid];
  __syncthreads();
  int mmv = mmflag[0] ? 1 : 0;
  int m0 = mmv ? 1 : 0, nm = mmv ? 3 : 1;
#ifdef HAVE_WMMA_F32X4
  {
    int wave = tid >> 5, lane = tid & 31;
    int half = lane >> 4, lrow = lane & 15;
    int n = wave * 16 + lrow;          // 8 waves cover N = 128
    v8f acc = {0.f, 0.f, 0.f, 0.f, 0.f, 0.f, 0.f, 0.f};
    for (int kk = 0; kk < C4N; kk += 4) {
      int ki = kk + 2 * half;
      v2f a = *(const v2f*)&At[lrow * 68 + ki];
      v2f bf;
      bf.x = w2g[ki * 128 + n];
      bf.y = w2g[(ki + 1) * 128 + n];
      acc = __builtin_amdgcn_wmma_f32_16x16x4_f32(false, a, false, bf, (short)0,
                                                  acc, false, false);
    }
    float bias = b2g[n >> 2];
    for (int r = 0; r < 8; ++r) {
      int mrow = r + 8 * half;
      G[mrow * 132 + n] = gelu_exact(acc[r] + bias);
    }
  }
#else
  {
    int n = tid & 127, rh = tid >> 7;  // 2 row-halves x 128 n
    float bias = b2g[n >> 2];
    for (int r8 = 0; r8 < 8; ++r8) {
      int row = rh * 8 + r8;
      float acc = bias;
      for (int i = 0; i < C4N; ++i) acc += At[row * 68 + i] * w2g[i * 128 + n];
      G[row * 132 + n] = gelu_exact(acc);
    }
  }
#endif
  __syncthreads();
  {
    int row = tid >> 4, sub = tid & 15;
    int mi = sub >> 2, tap = sub & 3;
    if (mi < nm) {
      float hsum = 0.f;
      const float* hp = &hyp_l[(m0 + mi) * C8N];
      for (int oc = 0; oc < C8N; ++oc) hsum += G[row * 132 + oc * 4 + tap] * hp[oc];
      int p2 = p2base + row;
      int h2 = p2 >> 7, w2i = p2 & 127;
      int P4 = (2 * h2 + (tap >> 1)) * 256 + 2 * w2i + (tap & 1);
      out[(size_t)b * nm * HW4N + (size_t)mi * HW4N + P4] = hsum;
    }
  }
}

// ---------------- K10: hyper MoE (per m,b), writes hyper_in + hyper_logits ----------------
__global__ __launch_bounds__(32) void k10_hyper_moe(
    const float* __restrict__ hs_g, const float* __restrict__ gw,
    const float* __restrict__ gb, const float* __restrict__ ew,
    const float* __restrict__ eb, const float* __restrict__ sw,
    const float* __restrict__ sb, const int* __restrict__ mmflag,
    float* __restrict__ hyperin, float* __restrict__ out) {
  int m = blockIdx.x, b = blockIdx.y, tid = threadIdx.x;
  __shared__ float hsrow[CN];
  __shared__ float l[EN];
  for (int k = 0; k < 8; ++k)
    hsrow[tid + k * 32] = hs_g[(b * TN + 1 + m) * CN + tid + k * 32];
  __syncthreads();
  int mmv = mmflag[0] ? 1 : 0;
  int nm = mmv ? 3 : 1;
  size_t hlbase = (size_t)BN * nm * HW4N + (size_t)BN * nm;
  if (tid < EN) {
    float acc = gb[m * EN + tid];
    for (int c = 0; c < CN; ++c) acc += hsrow[c] * gw[(m * CN + c) * EN + tid];
    l[tid] = acc;
    out[hlbase + (size_t)(m * BN + b) * EN + tid] = acc;
  }
  __syncthreads();
  float lv[4] = {l[0], l[1], l[2], l[3]};
  int e0 = 0; float b0 = lv[0];
  for (int e = 1; e < EN; ++e) if (lv[e] > b0) { b0 = lv[e]; e0 = e; }
  int e1 = -1; float b1v = -3.402823466e38f;
  for (int e = 0; e < EN; ++e) if (e != e0 && lv[e] > b1v) { b1v = lv[e]; e1 = e; }
  float p1 = expf(b1v - b0);
  float inv = 1.0f / (1.0f + p1);
  float wn0 = inv, wn1 = p1 * inv;
  int o = tid;
  float s0 = eb[(m * EN + e0) * C8N + o];
  float s1e = eb[(m * EN + e1) * C8N + o];
  float sh = sb[(m * SEN + 0) * C8N + o] + sb[(m * SEN + 1) * C8N + o];
  for (int c = 0; c < CN; ++c) {
    float h = hsrow[c];
    s0  += h * ew[((m * EN + e0) * CN + c) * C8N + o];
    s1e += h * ew[((m * EN + e1) * CN + c) * C8N + o];
    sh  += h * (sw[((m * SEN + 0) * CN + c) * C8N + o] +
                sw[((m * SEN + 1) * CN + c) * C8N + o]);
  }
  hyperin[(b * MN + m) * C8N + o] = wn0 * s0 + wn1 * s1e + sh;
}

// ---------------- K11: iou MoE (per b), writes iou_pred slice + iou_logits ----------------
__global__ __launch_bounds__(32) void k11_iou_moe(
    const float* __restrict__ hs_g, const float* __restrict__ gw,
    const float* __restrict__ gb, const float* __restrict__ ew,
    const float* __restrict__ eb, const float* __restrict__ sw,
    const float* __restrict__ sb, const int* __restrict__ mmflag,
    float* __restrict__ out) {
  int b = blockIdx.x, tid = threadIdx.x;
  __shared__ float hsrow[CN];
  __shared__ float l[EN];
  for (int k = 0; k < 8; ++k)
    hsrow[tid + k * 32] = hs_g[(b * TN + 0) * CN + tid + k * 32];
  __syncthreads();
  int mmv = mmflag[0] ? 1 : 0;
  int nm = mmv ? 3 : 1;
  size_t ipbase = (size_t)BN * nm * HW4N;
  size_t ilbase = ipbase + (size_t)BN * nm + (size_t)MN * BN * EN;
  if (tid < EN) {
    float acc = gb[tid];
    for (int c = 0; c < CN; ++c) acc += hsrow[c] * gw[c * EN + tid];
    l[tid] = acc;
    out[ilbase + (size_t)b * EN + tid] = acc;
  }
  __syncthreads();
  if (tid < MN) {
    float lv[4] = {l[0], l[1], l[2], l[3]};
    int e0 = 0; float b0 = lv[0];
    for (int e = 1; e < EN; ++e) if (lv[e] > b0) { b0 = lv[e]; e0 = e; }
    int e1 = -1; float b1v = -3.402823466e38f;
    for (int e = 0; e < EN; ++e) if (e != e0 && lv[e] > b1v) { b1v = lv[e]; e1 = e; }
    float p1 = expf(b1v - b0);
    float inv = 1.0f / (1.0f + p1);
    float wn0 = inv, wn1 = p1 * inv;
    int m4 = tid;
    float s0 = eb[e0 * MN + m4];
    float s1e = eb[e1 * MN + m4];
    float sh = sb[0 * MN + m4] + sb[1 * MN + m4];
    for (int c = 0; c < CN; ++c) {
      float h = hsrow[c];
      s0  += h * ew[(e0 * CN + c) * MN + m4];
      s1e += h * ew[(e1 * CN + c) * MN + m4];
      sh  += h * (sw[(0 * CN + c) * MN + m4] + sw[(1 * CN + c) * MN + m4]);
    }
    float pred = wn0 * s0 + wn1 * s1e + sh;
    if (mmv) {
      if (m4 >= 1) out[ipbase + (size_t)b * 3 + (m4 - 1)] = pred;
    } else {
      if (m4 == 0) out[ipbase + (size_t)b] = pred;
    }
  }
}

// ---------------- launcher ----------------
extern "C" void kernel_launch(void* const* d_in, const int* in_sizes, int n_in,
                              void* d_out, int out_size, void* d_ws, size_t ws_size,
                              hipStream_t stream) {
  const float* img    = (const float*)d_in[0];
  const float* pe     = (const float*)d_in[1];
  const float* sparse = (const float*)d_in[2];
  const float* dense  = (const float*)d_in[3];
  const int*   mmflag = (const int*)d_in[4];
  const float* iou_tok = (const float*)d_in[5];
  const float* mask_tok = (const float*)d_in[6];
  const float* wq = (const float*)d_in[7];
  const float* wk = (const float*)d_in[8];
  const float* wv = (const float*)d_in[9];
  const float* wq2 = (const float*)d_in[10];
  const float* wk2 = (const float*)d_in[11];
  const float* wv2 = (const float*)d_in[12];
  const float* up1_w = (const float*)d_in[13];
  const float* up1_b = (const float*)d_in[14];
  const float* ln_w = (const float*)d_in[15];
  const float* ln_b = (const float*)d_in[16];
  const float* up2_w = (const float*)d_in[17];
  const float* up2_b = (const float*)d_in[18];
  const float* hgw = (const float*)d_in[19];
  const float* hgb = (const float*)d_in[20];
  const float* hew = (const float*)d_in[21];
  const float* heb = (const float*)d_in[22];
  const float* hsw = (const float*)d_in[23];
  const float* hsb = (const float*)d_in[24];
  const float* igw = (const float*)d_in[25];
  const float* igb = (const float*)d_in[26];
  const float* iew = (const float*)d_in[27];
  const float* ieb = (const float*)d_in[28];
  const float* isw = (const float*)d_in[29];
  const float* isb = (const float*)d_in[30];
  float* out = (float*)d_out;
  float* ws = (float*)d_ws;

  float* tokens = ws + WS_TOK;
  float* qh     = ws + WS_QH;
  float* u_g    = ws + WS_U;
  float* hs     = ws + WS_HS;
  float* v2     = ws + WS_V2;
  float* w2t    = ws + WS_W2T;
  float* hypin  = ws + WS_HYP;
  float* attn   = ws + WS_ATT;
  float* src2   = ws + WS_BIGA;   // later reused as X2 (disjoint lifetimes)
  float* yb     = ws + WS_BIGB;
  float* X2     = ws + WS_BIGA;

  k0_tokens_qh<<<dim3(TN, BN), 256, 0, stream>>>(iou_tok, mask_tok, sparse, wq, wk,
                                                 tokens, qh);
  k1_scores1<<<dim3(16, BN), 256, 0, stream>>>(img, dense, pe, qh, attn);
  k2_softmax_rows<<<BN * TN, 256, 0, stream>>>(attn);
  k3_attn_kmem<<<BN * CN, 256, 0, stream>>>(img, dense, pe, attn, u_g);
  k4_hs<<<dim3(TN, BN), 256, 0, stream>>>(u_g, wv, tokens, hs);
  k10_hyper_moe<<<dim3(MN, BN), 32, 0, stream>>>(hs, hgw, hgb, hew, heb, hsw, hsb,
                                                 mmflag, hypin, out);
  k11_iou_moe<<<BN, 32, 0, stream>>>(hs, igw, igb, iew, ieb, isw, isb, mmflag, out);
  k5_proj2<<<BN, 256, 0, stream>>>(hs, wk2, wv2, wq2, v2, w2t);
  k6_attn2_src2<<<dim3(16, BN), 256, 0, stream>>>(img, dense, pe, w2t, v2, src2);
  k7_up1_gemm<<<BN * 256, 256, 0, stream>>>(src2, up1_w, yb);
  k8_ln_gelu<<<BN * HWN, 256, 0, stream>>>(yb, up1_b, ln_w, ln_b, X2);
  k9_up2_masks<<<BN * 1024, 256, 0, stream>>>(X2, up2_w, up2_b, hypin, mmflag, out);
}